// UMambaFirst_29850022708041
// MI455X (gfx1250) — compile-verified
//
#include <hip/hip_runtime.h>
#include <hip/hip_bf16.h>
#include <math.h>

typedef __attribute__((ext_vector_type(2))) float v2f;
typedef __attribute__((ext_vector_type(8))) float v8f;

#define LDIM   32768
#define CDIM   32
#define DINNER 64
#define DSTATE 16
#define DTRANK 2
#define DCONV  4
#define NCHUNK 128
#define CHLEN  (LDIM / NCHUNK)   // 256
#define LN_EPS 1e-5f

__device__ __forceinline__ float silu_f(float v) {
    return v / (1.0f + __expf(-v));
}

// ---------------------------------------------------------------------------
// WMMA f32 16x16x4 fragment coordinates (ISA 7.12.2):
//  A (16x4, MxK):  m = lane%16 ; v0 -> K = 2*(lane/16), v1 -> K+1
//  B (4x16, KxN):  n = lane%16 ; v0 -> K = 2*(lane/16), v1 -> K+1
//  C/D (16x16):    n = lane%16 ; vgpr v -> M = v + 8*(lane/16)
// ---------------------------------------------------------------------------

// Kernel 1: LayerNorm over channels + in_proj GEMM (L,32)x(32,128) -> xz(L,128)
__global__ void k1_ln_inproj(const float* __restrict__ x,
                             const float* __restrict__ ln_w,
                             const float* __restrict__ ln_b,
                             const float* __restrict__ Win,   // (128,32)
                             float* __restrict__ xz)          // (L,128)
{
    __shared__ float xn[16][CDIM + 1];
    const int l0 = blockIdx.x * 16;
    const int t  = threadIdx.x;                 // 0..255, 8 waves

    for (int e = t; e < 16 * CDIM; e += 256) {
        int r = e / CDIM, c = e % CDIM;
        xn[r][c] = x[(size_t)c * LDIM + (l0 + r)];
    }
    __syncthreads();

    if (t < 16) {
        float mu = 0.f;
        for (int c = 0; c < CDIM; ++c) mu += xn[t][c];
        mu *= (1.0f / CDIM);
        float var = 0.f;
        for (int c = 0; c < CDIM; ++c) { float d = xn[t][c] - mu; var += d * d; }
        var *= (1.0f / CDIM);
        float rinv = rsqrtf(var + LN_EPS);
        for (int c = 0; c < CDIM; ++c)
            xn[t][c] = (xn[t][c] - mu) * rinv * ln_w[c] + ln_b[c];
    }
    __syncthreads();

    const int wave = t >> 5, lane = t & 31;
    const int mL   = lane & 15;
    const int half = lane >> 4;
    const int n    = wave * 16 + mL;     // output column 0..127

    v8f acc = {0.f,0.f,0.f,0.f,0.f,0.f,0.f,0.f};
    #pragma unroll
    for (int kk = 0; kk < CDIM; kk += 4) {
        const int ka = kk + half * 2;
        v2f a, b;
        a.x = xn[mL][ka];     a.y = xn[mL][ka + 1];
        b.x = Win[n * CDIM + ka];
        b.y = Win[n * CDIM + ka + 1];
        acc = __builtin_amdgcn_wmma_f32_16x16x4_f32(
                  false, a, false, b, (short)0, acc, false, false);
    }

    #pragma unroll
    for (int v = 0; v < 8; ++v) {
        const int l = l0 + v + half * 8;
        xz[(size_t)l * (2 * DINNER) + n] = acc[v];   // branchless, lane-contig
    }
}

// Kernel 2: depthwise causal conv1d(4) + SiLU ; x branch is xz[:, 0:64]
__global__ void k2_conv_silu(const float* __restrict__ xz,    // (L,128)
                             const float* __restrict__ conv_w, // (64,1,4)
                             const float* __restrict__ conv_b,
                             float* __restrict__ xc)           // (L,64)
{
    const int idx = blockIdx.x * blockDim.x + threadIdx.x;
    if (idx >= LDIM * DINNER) return;
    const int l = idx / DINNER, d = idx % DINNER;
    float acc = conv_b[d];
    #pragma unroll
    for (int j = 0; j < DCONV; ++j) {
        const int ls = l - (DCONV - 1) + j;
        const float xv = (ls >= 0) ? xz[(size_t)ls * (2 * DINNER) + d] : 0.f;
        acc += conv_w[d * DCONV + j] * xv;
    }
    xc[idx] = silu_f(acc);
}

// Kernel 3: x_proj GEMM (L,64)x(64,34) -> dt(2),B(16),C(16); fused dt_proj+softplus
__global__ void k3_xproj_delta(const float* __restrict__ xc,
                               const float* __restrict__ Wx,   // (34,64)
                               const float* __restrict__ Wdt,  // (64,2)
                               const float* __restrict__ bdt,
                               float* __restrict__ Bm,         // (L,16)
                               float* __restrict__ Cm,         // (L,16)
                               float* __restrict__ delta)      // (L,64)
{
    __shared__ float sxc [16][DINNER + 1];
    __shared__ float sdbl[16][48 + 1];
    const int l0 = blockIdx.x * 16;
    const int t  = threadIdx.x;           // 0..127, 4 waves

    for (int e = t; e < 16 * DINNER; e += 128) {
        int r = e >> 6, c = e & 63;
        sxc[r][c] = xc[(size_t)(l0 + r) * DINNER + c];
    }
    __syncthreads();

    const int wave = t >> 5, lane = t & 31;
    if (wave < 3) {  // 3 N-tiles of 16 cover the 34 (padded to 48) outputs
        const int mL   = lane & 15;
        const int half = lane >> 4;
        const int n    = wave * 16 + mL;
        v8f acc = {0.f,0.f,0.f,0.f,0.f,0.f,0.f,0.f};
        #pragma unroll
        for (int kk = 0; kk < DINNER; kk += 4) {
            const int ka = kk + half * 2;
            v2f a, b;
            a.x = sxc[mL][ka]; a.y = sxc[mL][ka + 1];
            b.x = (n < 34) ? Wx[n * DINNER + ka]     : 0.f;
            b.y = (n < 34) ? Wx[n * DINNER + ka + 1] : 0.f;
            acc = __builtin_amdgcn_wmma_f32_16x16x4_f32(
                      false, a, false, b, (short)0, acc, false, false);
        }
        #pragma unroll
        for (int v = 0; v < 8; ++v) sdbl[v + half * 8][n] = acc[v];
    }
    __syncthreads();

    for (int e = t; e < 16 * DSTATE; e += 128) {
        int r = e / DSTATE, s = e % DSTATE;
        Bm[(size_t)(l0 + r) * DSTATE + s] = sdbl[r][DTRANK + s];
        Cm[(size_t)(l0 + r) * DSTATE + s] = sdbl[r][DTRANK + DSTATE + s];
    }
    for (int e = t; e < 16 * DINNER; e += 128) {
        int r = e >> 6, d = e & 63;
        float v = sdbl[r][0] * Wdt[d * DTRANK + 0]
                + sdbl[r][1] * Wdt[d * DTRANK + 1] + bdt[d];
        float sp = (v > 20.f) ? v : log1pf(__expf(v));
        delta[(size_t)(l0 + r) * DINNER + d] = sp;
    }
}

// Kernel 4a: per-chunk affine transfer (P = prod dA, Q = scan with h0=0)
__global__ void k4a_scan_chunk(const float* __restrict__ delta,
                               const float* __restrict__ Bm,
                               const float* __restrict__ xc,
                               const float* __restrict__ A_log,
                               float* __restrict__ P, float* __restrict__ Q)
{
    const int tid   = blockIdx.x * 256 + threadIdx.x;
    const int chunk = tid >> 10;
    const int r     = tid & 1023;
    const int d     = r >> 4, s = r & 15;
    const float a2 = -__expf(A_log[d * DSTATE + s]) * 1.44269504f; // log2(e) folded
    float p = 1.f, q = 0.f;
    const int lbeg = chunk * CHLEN;
    for (int i = 0; i < CHLEN; ++i) {
        const int l = lbeg + i;
        const float dl  = delta[(size_t)l * DINNER + d];
        const float dA  = exp2f(dl * a2);
        const float dBx = dl * Bm[(size_t)l * DSTATE + s]
                             * xc[(size_t)l * DINNER + d];
        p *= dA;
        q  = dA * q + dBx;
    }
    P[chunk * 1024 + r] = p;
    Q[chunk * 1024 + r] = q;
}

// Kernel 4b: sequential prefix over chunks (one thread per (d,s))
__global__ void k4b_prefix(const float* __restrict__ P,
                           const float* __restrict__ Q,
                           float* __restrict__ hstart)
{
    const int r = threadIdx.x;  // 1024 threads
    float h = 0.f;
    for (int c = 0; c < NCHUNK; ++c) {
        hstart[c * 1024 + r] = h;
        h = P[c * 1024 + r] * h + Q[c * 1024 + r];
    }
}

// Kernel 4c: rescan with correct h0; fuse y = <h,C> + xc*D, gate silu(z)
__global__ void k4c_scan_y(const float* __restrict__ delta,
                           const float* __restrict__ Bm,
                           const float* __restrict__ Cm,
                           const float* __restrict__ xc,
                           const float* __restrict__ zb,  // = xz + 64, stride 128
                           const float* __restrict__ A_log,
                           const float* __restrict__ Dp,
                           const float* __restrict__ hstart,
                           float* __restrict__ yw)   // (L,64)
{
    const int tid   = blockIdx.x * 256 + threadIdx.x;
    const int chunk = tid >> 10;
    const int r     = tid & 1023;
    const int d     = r >> 4, s = r & 15;
    const float a2  = -__expf(A_log[d * DSTATE + s]) * 1.44269504f;
    const float Dd  = Dp[d];
    float h = hstart[chunk * 1024 + r];
    const int lbeg = chunk * CHLEN;
    for (int i = 0; i < CHLEN; ++i) {
        const int l = lbeg + i;
        const float dl  = delta[(size_t)l * DINNER + d];
        const float xcv = xc[(size_t)l * DINNER + d];
        const float dA  = exp2f(dl * a2);
        const float dBx = dl * Bm[(size_t)l * DSTATE + s] * xcv;
        h = dA * h + dBx;
        float yc = h * Cm[(size_t)l * DSTATE + s];
        #pragma unroll
        for (int off = 8; off >= 1; off >>= 1) yc += __shfl_xor(yc, off, 32);
        if (s == 0) {
            const float zz = zb[(size_t)l * (2 * DINNER) + d];
            yw[(size_t)l * DINNER + d] = (yc + xcv * Dd) * silu_f(zz);
        }
    }
}

// Kernel 5: out_proj GEMM (L,64)x(64,32), write transposed to (C,L)
__global__ void k5_outproj(const float* __restrict__ yw,
                           const float* __restrict__ Wout,  // (32,64)
                           float* __restrict__ out)          // (32,L)
{
    __shared__ float syw[64][DINNER + 1];     // 64-row tile of yw
    const int l0 = blockIdx.x * 64;
    const int t  = threadIdx.x, wave = t >> 5, lane = t & 31;

    for (int e = t; e < 64 * DINNER; e += 256) {
        int r = e >> 6, c = e & 63;
        syw[r][c] = yw[(size_t)(l0 + r) * DINNER + c];
    }
    __syncthreads();

    const int mt = wave >> 1, nt = wave & 1;    // 4 M-tiles x 2 N-tiles
    const int mL = lane & 15, half = lane >> 4;
    const int n  = nt * 16 + mL;
    const int mrow = mt * 16 + mL;

    v8f acc = {0.f,0.f,0.f,0.f,0.f,0.f,0.f,0.f};
    #pragma unroll
    for (int kk = 0; kk < DINNER; kk += 4) {
        const int ka = kk + half * 2;
        v2f a, b;
        a.x = syw[mrow][ka];  a.y = syw[mrow][ka + 1];
        b.x = Wout[n * DINNER + ka];
        b.y = Wout[n * DINNER + ka + 1];
        acc = __builtin_amdgcn_wmma_f32_16x16x4_f32(
                  false, a, false, b, (short)0, acc, false, false);
    }
    #pragma unroll
    for (int v = 0; v < 8; ++v) {
        const int l = l0 + mt * 16 + v + half * 8;
        out[(size_t)n * LDIM + l] = acc[v];
    }
}

extern "C" void kernel_launch(void* const* d_in, const int* in_sizes, int n_in,
                              void* d_out, int out_size, void* d_ws, size_t ws_size,
                              hipStream_t stream) {
    (void)in_sizes; (void)n_in; (void)out_size; (void)ws_size;
    const float* x        = (const float*)d_in[0];
    const float* ln_w     = (const float*)d_in[1];
    const float* ln_b     = (const float*)d_in[2];
    const float* in_proj  = (const float*)d_in[3];
    const float* conv_w   = (const float*)d_in[4];
    const float* conv_b   = (const float*)d_in[5];
    const float* x_proj   = (const float*)d_in[6];
    const float* dt_w     = (const float*)d_in[7];
    const float* dt_b     = (const float*)d_in[8];
    const float* A_log    = (const float*)d_in[9];
    const float* D_param  = (const float*)d_in[10];
    const float* out_proj = (const float*)d_in[11];
    float* out = (float*)d_out;

    const size_t N64 = (size_t)LDIM * DINNER;   // 2,097,152
    const size_t N16 = (size_t)LDIM * DSTATE;   //   524,288
    const size_t NC  = (size_t)NCHUNK * 1024;   //   131,072
    float* ws     = (float*)d_ws;
    float* xz     = ws;                  // (L,128): [:,0:64]=x branch, [:,64:128]=z
    float* xc     = ws + 2 * N64;
    float* delta  = ws + 3 * N64;
    float* yw     = ws + 4 * N64;
    float* Bm     = ws + 5 * N64;
    float* Cm     = Bm + N16;
    float* P      = Cm + N16;
    float* Q      = P + NC;
    float* hstart = Q + NC;

    k1_ln_inproj<<<LDIM / 16, 256, 0, stream>>>(x, ln_w, ln_b, in_proj, xz);
    k2_conv_silu<<<(LDIM * DINNER) / 256, 256, 0, stream>>>(xz, conv_w, conv_b, xc);
    k3_xproj_delta<<<LDIM / 16, 128, 0, stream>>>(xc, x_proj, dt_w, dt_b, Bm, Cm, delta);
    k4a_scan_chunk<<<(NCHUNK * 1024) / 256, 256, 0, stream>>>(delta, Bm, xc, A_log, P, Q);
    k4b_prefix<<<1, 1024, 0, stream>>>(P, Q, hstart);
    k4c_scan_y<<<(NCHUNK * 1024) / 256, 256, 0, stream>>>(delta, Bm, Cm, xc, xz + DINNER,
                                                          A_log, D_param, hstart, yw);
    k5_outproj<<<LDIM / 64, 256, 0, stream>>>(yw, out_proj, out);
}